// KBLN_10179072491913
// MI455X (gfx1250) — compile-verified
//
#include <hip/hip_runtime.h>

typedef __attribute__((ext_vector_type(2))) float v2f;
typedef __attribute__((ext_vector_type(4))) float v4f;
typedef __attribute__((ext_vector_type(8))) float v8f;

#define B_ 256
#define E_ 15000
#define D_ 200
#define L_ 64
#define QS_STRIDE 204   /* 200 padded: kills LDS bank conflicts for A-frag reads */
#define LS_STRIDE 68    /* 64 padded:  row stride 4 banks -> conflict-free b128 reads */

// ---------------------------------------------------------------------------
// Prep: gather per-row data once.
//   Q[b,d]  = emb_e[e1[b],d] * emb_rel[rel[b],d]
//   A[b,l]  = lit[e1[b],l] - c[l]
//   W[b,l]  = nf_weights[rel[b],l]
//   IV[l]   = -log2(e)/var[l]   (so inner loop is a single v_exp_f32)
// ---------------------------------------------------------------------------
__global__ __launch_bounds__(64) void KBLN_prep(
    const float* __restrict__ emb_e, const float* __restrict__ emb_rel,
    const float* __restrict__ nf,    const float* __restrict__ lit,
    const float* __restrict__ c,     const float* __restrict__ var,
    const int*   __restrict__ e1,    const int*   __restrict__ rel,
    float* __restrict__ Q, float* __restrict__ A,
    float* __restrict__ W, float* __restrict__ IV)
{
    const int b = blockIdx.x;
    const int t = threadIdx.x;
    const int ie = e1[b];
    const int ir = rel[b];
    for (int d = t; d < D_; d += 64)
        Q[b * D_ + d] = emb_e[(size_t)ie * D_ + d] * emb_rel[(size_t)ir * D_ + d];
    A[b * L_ + t] = lit[(size_t)ie * L_ + t] - c[t];
    W[b * L_ + t] = nf[(size_t)ir * L_ + t];
    if (b == 0) IV[t] = -1.44269504088896341f / var[t];
}

// ---------------------------------------------------------------------------
// Main: each 256-thread block = 8 waves; wave w owns a 16(b) x 16(e) tile.
// Block tile: 16 b-rows x 128 e-cols. GEMM via v_wmma_f32_16x16x4_f32,
// then RBF sum (64 exp per output) entirely from LDS, fused sigmoid.
// ---------------------------------------------------------------------------
__global__ __launch_bounds__(256) void KBLN_main(
    const float* __restrict__ emb_e, const float* __restrict__ lit,
    const float* __restrict__ Q,     const float* __restrict__ A,
    const float* __restrict__ W,     const float* __restrict__ IV,
    float* __restrict__ out)
{
    __shared__ __attribute__((aligned(16))) float Qs[16 * QS_STRIDE];
    __shared__ __attribute__((aligned(16))) float As[16 * L_];
    __shared__ __attribute__((aligned(16))) float Ws[16 * L_];
    __shared__ __attribute__((aligned(16))) float IVs[L_];
    __shared__ __attribute__((aligned(16))) float Ls[8 * 16 * LS_STRIDE];

    const int tid  = threadIdx.x;
    const int wave = tid >> 5;
    const int lane = tid & 31;
    const int b0   = blockIdx.y * 16;
    const int e0   = blockIdx.x * 128 + wave * 16;

    // ---- stage Q tile (shared by all 8 waves) ----
    for (int idx = tid; idx < 16 * D_; idx += 256) {
        int r = idx / D_, cc = idx % D_;
        Qs[r * QS_STRIDE + cc] = Q[(b0 + r) * D_ + cc];
    }
    // ---- stage A / W tiles (shared) ----
    for (int idx = tid; idx < 16 * L_; idx += 256) {
        As[idx] = A[b0 * L_ + idx];
        Ws[idx] = W[b0 * L_ + idx];
    }
    if (tid < L_) IVs[tid] = IV[tid];
    // ---- stage per-wave lit tile (16 e-rows x 64), clamped at E ----
    {
        float* Lw = &Ls[wave * 16 * LS_STRIDE];
        #pragma unroll 4
        for (int t = 0; t < 32; ++t) {
            int idx = t * 32 + lane;
            int r = idx >> 6, cc = idx & 63;
            int ec = e0 + r; if (ec >= E_) ec = E_ - 1;
            Lw[r * LS_STRIDE + cc] = lit[(size_t)ec * L_ + cc];
        }
    }
    __syncthreads();

    const int half = lane >> 4;   // 0: lanes 0-15, 1: lanes 16-31
    const int mn   = lane & 15;   // row (A) / col (B) index
    const int kb   = half * 2;    // K sub-offset within 16x16x4 fragment

    // ---- GEMM: acc(16x16) += Qtile(16x4) * emb_e^T(4x16), over D=200 ----
    v8f acc = {};
    int ecol = e0 + mn;
    int ecolc = ecol < E_ ? ecol : (E_ - 1);
    const float* __restrict__ Brow = emb_e + (size_t)ecolc * D_;
    const float* Arow = &Qs[mn * QS_STRIDE];
    for (int kk = 0; kk < D_; kk += 4) {
        v2f af = *(const v2f*)(Arow + kk + kb);   // A[m][kk+kb .. +1]
        v2f bf = *(const v2f*)(Brow + kk + kb);   // B[kk+kb .. +1][n]
        acc = __builtin_amdgcn_wmma_f32_16x16x4_f32(
            /*neg_a=*/false, af, /*neg_b=*/false, bf,
            /*c_mod=*/(short)0, acc, /*reuse_a=*/false, /*reuse_b=*/false);
    }

    // ---- RBF literal score + sigmoid, from LDS only ----
    const float* Lrow = &Ls[(wave * 16 + mn) * LS_STRIDE];
    const bool evalid = (e0 + mn) < E_;
    float* outp = out + (size_t)b0 * E_ + (e0 + mn);
    const float LOG2E = 1.44269504088896341f;

    #pragma unroll
    for (int i = 0; i < 8; ++i) {
        const int row = i + half * 8;             // b-row in tile per C layout
        const float* Ar = &As[row * L_];
        const float* Wr = &Ws[row * L_];
        float s = 0.0f;
        for (int lq = 0; lq < L_; lq += 4) {
            v4f a4 = *(const v4f*)(Ar + lq);
            v4f w4 = *(const v4f*)(Wr + lq);
            v4f e4 = *(const v4f*)(Lrow + lq);
            v4f i4 = *(const v4f*)(IVs + lq);
            #pragma unroll
            for (int j = 0; j < 4; ++j) {
                float t = a4[j] - e4[j];
                s += __builtin_amdgcn_exp2f(t * t * i4[j]) * w4[j];
            }
        }
        float x = acc[i] + s;
        float sg = 1.0f / (1.0f + __builtin_amdgcn_exp2f(-LOG2E * x));
        if (evalid) outp[(size_t)row * E_] = sg;
    }
}

extern "C" void kernel_launch(void* const* d_in, const int* in_sizes, int n_in,
                              void* d_out, int out_size, void* d_ws, size_t ws_size,
                              hipStream_t stream) {
    const float* emb_e   = (const float*)d_in[0];
    const float* emb_rel = (const float*)d_in[1];
    const float* nf      = (const float*)d_in[2];
    const float* lit     = (const float*)d_in[3];
    const float* c       = (const float*)d_in[4];
    const float* var     = (const float*)d_in[5];
    const int*   e1      = (const int*)d_in[6];
    const int*   rel     = (const int*)d_in[7];
    float* out = (float*)d_out;

    float* Q  = (float*)d_ws;        // 256*200
    float* A  = Q + B_ * D_;         // 256*64
    float* W  = A + B_ * L_;         // 256*64
    float* IV = W + B_ * L_;         // 64

    KBLN_prep<<<B_, 64, 0, stream>>>(emb_e, emb_rel, nf, lit, c, var, e1, rel,
                                     Q, A, W, IV);
    dim3 grid((E_ + 127) / 128, B_ / 16);
    KBLN_main<<<grid, 256, 0, stream>>>(emb_e, lit, Q, A, W, IV, out);
}